// GATEncoder_82703890252075
// MI455X (gfx1250) — compile-verified
//
#include <hip/hip_runtime.h>
#include <math.h>

#define N_NODES 50000
#define N_EDGES 800000
#define E_TOT   (N_EDGES + N_NODES)   // self-loops appended logically
#define NEG_SLOPE 0.2f

typedef float v2f __attribute__((ext_vector_type(2)));
typedef float v8f __attribute__((ext_vector_type(8)));

__device__ __forceinline__ float leaky(float a) { return a > 0.0f ? a : NEG_SLOPE * a; }

__device__ __forceinline__ void atomicMaxF(float* addr, float val) {
    unsigned int* ua = (unsigned int*)addr;
    unsigned int old = *ua;
    while (__uint_as_float(old) < val) {
        unsigned int assumed = old;
        old = atomicCAS(ua, assumed, __float_as_uint(val));
        if (old == assumed) break;
    }
}

// ---------------------------------------------------------------- fills
__global__ void k_fill(float* __restrict__ p, float v, int n) {
    int i = blockIdx.x * blockDim.x + threadIdx.x;
    if (i < n) p[i] = v;
}

// ------------------------------------------------- layer1: h = x@W1, logits
// thread per (node, head): K=2, 32 channels each
__global__ void k_l1_feat(const float* __restrict__ x, const float* __restrict__ W1,
                          const float* __restrict__ as1, const float* __restrict__ ad1,
                          float* __restrict__ h1, float* __restrict__ asrc, float* __restrict__ adst) {
    int i = blockIdx.x * blockDim.x + threadIdx.x;
    if (i >= N_NODES * 4) return;
    int n = i >> 2, hh = i & 3;
    float x0 = x[n * 2 + 0], x1 = x[n * 2 + 1];
    float s = 0.f, d = 0.f;
    float* hp = h1 + (size_t)n * 128 + hh * 32;
    for (int c = 0; c < 32; ++c) {
        float v = fmaf(x0, W1[hh * 32 + c], x1 * W1[128 + hh * 32 + c]);
        hp[c] = v;
        s = fmaf(v, as1[hh * 32 + c], s);
        d = fmaf(v, ad1[hh * 32 + c], d);
    }
    asrc[i] = s;
    adst[i] = d;
}

// ------------------------------------------------- edge pass A: segment max (4 heads)
__global__ void k_edge_max1(const int* __restrict__ src, const int* __restrict__ dst,
                            const float* __restrict__ asrc, const float* __restrict__ adst,
                            float* __restrict__ amax) {
    int e = blockIdx.x * blockDim.x + threadIdx.x;
    if (e >= E_TOT) return;
    int s, d;
    if (e < N_EDGES) { s = src[e]; d = dst[e]; } else { s = d = e - N_EDGES; }
    for (int hh = 0; hh < 4; ++hh) {
        float a = leaky(asrc[s * 4 + hh] + adst[d * 4 + hh]);
        atomicMaxF(&amax[d * 4 + hh], a);
    }
}

// ------------------------------------------------- edge pass B: exp + denom
__global__ void k_edge_den1(const int* __restrict__ src, const int* __restrict__ dst,
                            const float* __restrict__ asrc, const float* __restrict__ adst,
                            const float* __restrict__ amax, float* __restrict__ den) {
    int e = blockIdx.x * blockDim.x + threadIdx.x;
    if (e >= E_TOT) return;
    int s, d;
    if (e < N_EDGES) { s = src[e]; d = dst[e]; } else { s = d = e - N_EDGES; }
    for (int hh = 0; hh < 4; ++hh) {
        float a = leaky(asrc[s * 4 + hh] + adst[d * 4 + hh]);
        atomicAdd(&den[d * 4 + hh], __expf(a - amax[d * 4 + hh]));
    }
}

// ------------------------------------------------- edge pass C: message scatter
// thread per (edge, head): 32 channels
__global__ void k_edge_agg1(const int* __restrict__ src, const int* __restrict__ dst,
                            const float* __restrict__ asrc, const float* __restrict__ adst,
                            const float* __restrict__ amax, const float* __restrict__ den,
                            const float* __restrict__ h1, float* __restrict__ agg) {
    int i = blockIdx.x * blockDim.x + threadIdx.x;
    if (i >= E_TOT * 4) return;
    int e = i >> 2, hh = i & 3;
    int s, d;
    if (e < N_EDGES) { s = src[e]; d = dst[e]; } else { s = d = e - N_EDGES; }
    float a = leaky(asrc[s * 4 + hh] + adst[d * 4 + hh]);
    float coef = __expf(a - amax[d * 4 + hh]) / (den[d * 4 + hh] + 1e-16f);
    const float* hs = h1 + (size_t)s * 128 + hh * 32;
    float* od = agg + (size_t)d * 128 + hh * 32;
    for (int c = 0; c < 32; ++c) atomicAdd(&od[c], hs[c] * coef);
}

// ------------------------------------------------- bias + ReLU -> h1relu
__global__ void k_relu_bias(const float* __restrict__ agg, const float* __restrict__ b1,
                            float* __restrict__ out) {
    int i = blockIdx.x * blockDim.x + threadIdx.x;
    if (i >= N_NODES * 128) return;
    float v = agg[i] + b1[i & 127];
    out[i] = v > 0.f ? v : 0.f;
}

// ------------------------------------------------- layer2 GEMM: h2 = h1relu @ W2
// fp32 WMMA 16x16x4. Block = 256 thr = 8 waves; each wave owns a 16-row strip
// and all 8 column tiles (16x128 of output). W2 staged in 64KB LDS.
__global__ __launch_bounds__(256) void k_gemm_wmma(const float* __restrict__ A,
                                                   const float* __restrict__ W,
                                                   float* __restrict__ C) {
    __shared__ float lw[128 * 128];  // 64 KB (WGP has 320 KB)
    int tid = threadIdx.x;
    for (int i = tid; i < 128 * 128 / 4; i += 256)
        ((float4*)lw)[i] = ((const float4*)W)[i];
    __syncthreads();

    int wave = tid >> 5;
    int lane = tid & 31;
    int l16  = lane & 15;
    int hi   = lane >> 4;                    // lane half selects K pair / M half
    int row0 = (blockIdx.x * 8 + wave) * 16;

    v8f acc[8];
    for (int t = 0; t < 8; ++t) acc[t] = (v8f)(0.0f);

    int ar = row0 + l16;
    if (ar > N_NODES - 1) ar = N_NODES - 1;  // clamp pad rows (stores guarded)
    const float* arow = A + (size_t)ar * 128;

    for (int k4 = 0; k4 < 128; k4 += 4) {
        int kk = k4 + 2 * hi;
        // A 16x4 fragment: lane<16 holds K={k4,k4+1}, lane>=16 holds K={k4+2,k4+3}
        v2f a;
        a.x = arow[kk + 0];
        a.y = arow[kk + 1];
        for (int t = 0; t < 8; ++t) {
            // B 4x16 fragment (mirrors A): vgpr0/1 = rows kk,kk+1; cols t*16+l16
            v2f b;
            b.x = lw[(kk + 0) * 128 + t * 16 + l16];
            b.y = lw[(kk + 1) * 128 + t * 16 + l16];
            acc[t] = __builtin_amdgcn_wmma_f32_16x16x4_f32(
                false, a, false, b, (short)0, acc[t], false, false);
        }
    }

    // D layout: VGPR r, lane<16 -> M=r, lane>=16 -> M=r+8; N = t*16+l16
    for (int t = 0; t < 8; ++t)
        for (int r = 0; r < 8; ++r) {
            int row = row0 + r + 8 * hi;
            if (row < N_NODES) C[(size_t)row * 128 + t * 16 + l16] = acc[t][r];
        }
}

// ------------------------------------------------- layer2 attention logits
__global__ void k_attn2(const float* __restrict__ h2, const float* __restrict__ as2,
                        const float* __restrict__ ad2, float* __restrict__ asrc,
                        float* __restrict__ adst) {
    int n = blockIdx.x * blockDim.x + threadIdx.x;
    if (n >= N_NODES) return;
    const float* hp = h2 + (size_t)n * 128;
    float s = 0.f, d = 0.f;
    for (int c = 0; c < 128; ++c) {
        float v = hp[c];
        s = fmaf(v, as2[c], s);
        d = fmaf(v, ad2[c], d);
    }
    asrc[n] = s;
    adst[n] = d;
}

// ------------------------------------------------- layer2 edge passes (1 head)
__global__ void k_edge_max2(const int* __restrict__ src, const int* __restrict__ dst,
                            const float* __restrict__ asrc, const float* __restrict__ adst,
                            float* __restrict__ amax) {
    int e = blockIdx.x * blockDim.x + threadIdx.x;
    if (e >= E_TOT) return;
    int s, d;
    if (e < N_EDGES) { s = src[e]; d = dst[e]; } else { s = d = e - N_EDGES; }
    atomicMaxF(&amax[d], leaky(asrc[s] + adst[d]));
}

__global__ void k_edge_den2(const int* __restrict__ src, const int* __restrict__ dst,
                            const float* __restrict__ asrc, const float* __restrict__ adst,
                            const float* __restrict__ amax, float* __restrict__ den) {
    int e = blockIdx.x * blockDim.x + threadIdx.x;
    if (e >= E_TOT) return;
    int s, d;
    if (e < N_EDGES) { s = src[e]; d = dst[e]; } else { s = d = e - N_EDGES; }
    float a = leaky(asrc[s] + adst[d]);
    atomicAdd(&den[d], __expf(a - amax[d]));
}

// thread per (edge, quarter): 32 channels each
__global__ void k_edge_agg2(const int* __restrict__ src, const int* __restrict__ dst,
                            const float* __restrict__ asrc, const float* __restrict__ adst,
                            const float* __restrict__ amax, const float* __restrict__ den,
                            const float* __restrict__ h2, float* __restrict__ agg) {
    int i = blockIdx.x * blockDim.x + threadIdx.x;
    if (i >= E_TOT * 4) return;
    int e = i >> 2, q = i & 3;
    int s, d;
    if (e < N_EDGES) { s = src[e]; d = dst[e]; } else { s = d = e - N_EDGES; }
    float a = leaky(asrc[s] + adst[d]);
    float coef = __expf(a - amax[d]) / (den[d] + 1e-16f);
    const float* hs = h2 + (size_t)s * 128 + q * 32;
    float* od = agg + (size_t)d * 128 + q * 32;
    for (int c = 0; c < 32; ++c) atomicAdd(&od[c], hs[c] * coef);
}

// ------------------------------------------------- global mean pool + b2
__global__ __launch_bounds__(256) void k_mean(const float* __restrict__ agg,
                                              const float* __restrict__ b2,
                                              float* __restrict__ out) {
    int c = blockIdx.x;  // 0..127
    float s = 0.f;
    for (int n = threadIdx.x; n < N_NODES; n += 256) s += agg[(size_t)n * 128 + c];
    __shared__ float red[256];
    red[threadIdx.x] = s;
    __syncthreads();
    for (int w = 128; w > 0; w >>= 1) {
        if (threadIdx.x < w) red[threadIdx.x] += red[threadIdx.x + w];
        __syncthreads();
    }
    if (threadIdx.x == 0) out[c] = red[0] * (1.0f / (float)N_NODES) + b2[c];
}

// ================================================================ launcher
extern "C" void kernel_launch(void* const* d_in, const int* in_sizes, int n_in,
                              void* d_out, int out_size, void* d_ws, size_t ws_size,
                              hipStream_t stream) {
    const float* x    = (const float*)d_in[0];
    const int*   ei   = (const int*)d_in[1];
    const float* W1   = (const float*)d_in[2];
    const float* as1  = (const float*)d_in[3];
    const float* ad1  = (const float*)d_in[4];
    const float* b1   = (const float*)d_in[5];
    const float* W2   = (const float*)d_in[6];
    const float* as2  = (const float*)d_in[7];
    const float* ad2  = (const float*)d_in[8];
    const float* b2   = (const float*)d_in[9];
    float*       out  = (float*)d_out;

    const int* src = ei;            // edge_index[0]
    const int* dst = ei + N_EDGES;  // edge_index[1]

    float* ws    = (float*)d_ws;
    float* bufA  = ws;                                 // N*128: h1, then h1relu, then out2-acc
    float* bufB  = bufA + (size_t)N_NODES * 128;       // N*128: agg1, then h2
    float* asrc1 = bufB + (size_t)N_NODES * 128;       // N*4
    float* adst1 = asrc1 + N_NODES * 4;
    float* amax1 = adst1 + N_NODES * 4;
    float* den1  = amax1 + N_NODES * 4;
    float* asrc2 = den1 + N_NODES * 4;                 // N
    float* adst2 = asrc2 + N_NODES;
    float* amax2 = adst2 + N_NODES;
    float* den2  = amax2 + N_NODES;

    const int B = 256;
    auto blocks = [](long long n) { return (unsigned)((n + 255) / 256); };
    const float NEG_INF = -INFINITY;

    // ---- layer 1
    k_l1_feat<<<blocks((long long)N_NODES * 4), B, 0, stream>>>(x, W1, as1, ad1, bufA, asrc1, adst1);
    k_fill<<<blocks(N_NODES * 4), B, 0, stream>>>(amax1, NEG_INF, N_NODES * 4);
    k_fill<<<blocks(N_NODES * 4), B, 0, stream>>>(den1, 0.0f, N_NODES * 4);
    k_fill<<<blocks((long long)N_NODES * 128), B, 0, stream>>>(bufB, 0.0f, N_NODES * 128);
    k_edge_max1<<<blocks(E_TOT), B, 0, stream>>>(src, dst, asrc1, adst1, amax1);
    k_edge_den1<<<blocks(E_TOT), B, 0, stream>>>(src, dst, asrc1, adst1, amax1, den1);
    k_edge_agg1<<<blocks((long long)E_TOT * 4), B, 0, stream>>>(src, dst, asrc1, adst1, amax1, den1, bufA, bufB);
    k_relu_bias<<<blocks((long long)N_NODES * 128), B, 0, stream>>>(bufB, b1, bufA);

    // ---- layer 2 dense projection via fp32 WMMA (h2 -> bufB)
    k_gemm_wmma<<<(N_NODES + 127) / 128, 256, 0, stream>>>(bufA, W2, bufB);

    // ---- layer 2 attention
    k_attn2<<<blocks(N_NODES), B, 0, stream>>>(bufB, as2, ad2, asrc2, adst2);
    k_fill<<<blocks(N_NODES), B, 0, stream>>>(amax2, NEG_INF, N_NODES);
    k_fill<<<blocks(N_NODES), B, 0, stream>>>(den2, 0.0f, N_NODES);
    k_fill<<<blocks((long long)N_NODES * 128), B, 0, stream>>>(bufA, 0.0f, N_NODES * 128);
    k_edge_max2<<<blocks(E_TOT), B, 0, stream>>>(src, dst, asrc2, adst2, amax2);
    k_edge_den2<<<blocks(E_TOT), B, 0, stream>>>(src, dst, asrc2, adst2, amax2, den2);
    k_edge_agg2<<<blocks((long long)E_TOT * 4), B, 0, stream>>>(src, dst, asrc2, adst2, amax2, den2, bufB, bufA);

    // ---- global mean pool + bias
    k_mean<<<128, 256, 0, stream>>>(bufA, b2, out);
}